// Net_2791728742835
// MI455X (gfx1250) — compile-verified
//
#include <hip/hip_runtime.h>

typedef __attribute__((ext_vector_type(2))) float v2f;
typedef __attribute__((ext_vector_type(8))) float v8f;

#define HIDF 64
#define NEG_SLOPE 0.01f

__device__ __forceinline__ float leaky(float x) {
    return x >= 0.f ? x : NEG_SLOPE * x;
}

// ---------------- degree / norm ----------------
__global__ void k_init_deg(float* __restrict__ deg, int n) {
    int i = blockIdx.x * blockDim.x + threadIdx.x;
    if (i < n) deg[i] = 1.0f;  // self-loop contributes 1
}

__global__ void k_count_deg(const int* __restrict__ dst, float* __restrict__ deg, int n_edges) {
    int e = blockIdx.x * blockDim.x + threadIdx.x;
    if (e < n_edges) atomicAdd(&deg[dst[e]], 1.0f);
}

__global__ void k_rsqrt_inplace(float* __restrict__ deg, int n) {
    int i = blockIdx.x * blockDim.x + threadIdx.x;
    if (i < n) {
        float d = deg[i];
        deg[i] = (d > 0.f) ? rsqrtf(d) : 0.f;
    }
}

__global__ void k_zero(float* __restrict__ p, int n) {
    int i = blockIdx.x * blockDim.x + threadIdx.x;
    if (i < n) p[i] = 0.f;
}

// ---------------- WMMA f32 GEMM ----------------
// out[n_rows x NCOLS] = A[n_rows x K] * B[K x NCOLS] (+ bias)
// v_wmma_f32_16x16x4_f32; one wave owns a 16-row x NCOLS tile and grid-strides
// over row tiles. B fragments are register-resident across the whole loop.
// A frag (lane L): m = L%16, k = 4*ks + 2*(L/16) + {0,1}
// B frag (lane L): n = L%16, same k mapping
// C/D  (lane L, vgpr r): row = r + 8*(L/16), col = L%16
template <int K, int NCOLS>
__global__ void k_gemm_wmma(const float* __restrict__ A, const float* __restrict__ B,
                            const float* __restrict__ bias, int use_bias,
                            float* __restrict__ out, int n_rows) {
    constexpr int KSTEPS = (K + 3) / 4;
    constexpr int NT = (NCOLS + 15) / 16;
    const int nwaves = (int)((gridDim.x * blockDim.x) >> 5);
    const int wave0 = (int)((blockIdx.x * blockDim.x + threadIdx.x) >> 5);
    const int lane = (int)(threadIdx.x & 31);
    const int m = lane & 15;
    const int half = lane >> 4;

    // ---- preload B fragments (branchless edge handling via clamp + select) ----
    v2f bf[KSTEPS][NT];
    float biasv[NT];
#pragma unroll
    for (int ks = 0; ks < KSTEPS; ++ks) {
        const int kb = ks * 4 + 2 * half;
        const int kb0 = (kb + 1 < K) ? kb : 0;  // keep the pair in-bounds
#pragma unroll
        for (int nt = 0; nt < NT; ++nt) {
            const int n = nt * 16 + m;
            const int nc = (NCOLS % 16 == 0) ? n : ((n < NCOLS) ? n : (NCOLS - 1));
            const float bx = B[(size_t)kb0 * NCOLS + nc];
            const float by = B[(size_t)(kb0 + 1) * NCOLS + nc];
            const bool nok = (NCOLS % 16 == 0) || (n < NCOLS);
            v2f b;
            b.x = ((kb < K) && nok) ? bx : 0.f;
            b.y = ((kb + 1 < K) && nok) ? by : 0.f;
            bf[ks][nt] = b;
        }
    }
#pragma unroll
    for (int nt = 0; nt < NT; ++nt) {
        const int n = nt * 16 + m;
        const int nc = (NCOLS % 16 == 0) ? n : ((n < NCOLS) ? n : (NCOLS - 1));
        biasv[nt] = use_bias ? bias[nc] : 0.f;
    }

    // ---- grid-stride over 16-row tiles (loop bound is wave-uniform: EXEC==all-1s at WMMA) ----
    for (int tile = wave0; tile * 16 < n_rows; tile += nwaves) {
        const int row0 = tile << 4;
        const float* __restrict__ arow = A + (size_t)(row0 + m) * (size_t)K;

        v8f acc[NT];
#pragma unroll
        for (int nt = 0; nt < NT; ++nt) acc[nt] = (v8f){0.f, 0.f, 0.f, 0.f, 0.f, 0.f, 0.f, 0.f};

#pragma unroll
        for (int ks = 0; ks < KSTEPS; ++ks) {
            const int kb = ks * 4 + 2 * half;
            v2f a;
            if constexpr (K % 4 == 0) {
                a = *(const v2f*)(arow + kb);  // 8B-aligned pair -> global_load_b64
            } else {
                const int kb0 = (kb + 1 < K) ? kb : 0;  // clamp pair in-bounds
                const v2f t = *(const v2f*)(arow + kb0);
                a.x = (kb < K) ? t.x : 0.f;
                a.y = (kb + 1 < K) ? t.y : 0.f;
            }
#pragma unroll
            for (int nt = 0; nt < NT; ++nt) {
                acc[nt] = __builtin_amdgcn_wmma_f32_16x16x4_f32(
                    false, a, false, bf[ks][nt], (short)0, acc[nt], false, false);
            }
        }

#pragma unroll
        for (int nt = 0; nt < NT; ++nt) {
            const int n = nt * 16 + m;
            if ((NCOLS % 16 == 0) || (n < NCOLS)) {
#pragma unroll
                for (int r = 0; r < 8; ++r) {
                    const int row = row0 + r + 8 * half;
                    out[(size_t)row * NCOLS + n] = acc[nt][r] + biasv[nt];
                }
            }
        }
    }
}

// ---------------- edge scatter-add ----------------
// 16 threads per edge, each handling 4 features (float4 gather + 4 f32 atomics).
__global__ void k_aggregate(const int* __restrict__ src, const int* __restrict__ dst,
                            const float* __restrict__ dis, const float* __restrict__ h,
                            float* __restrict__ agg, int n_edges) {
    unsigned t = blockIdx.x * blockDim.x + threadIdx.x;
    int e = (int)(t >> 4);
    if (e >= n_edges) return;
    int f4 = (int)(t & 15u) << 2;
    int s = src[e];
    int d = dst[e];
    float norm = dis[s] * dis[d];
    const float4 hv = *reinterpret_cast<const float4*>(h + (size_t)s * HIDF + f4);
    float* o = agg + (size_t)d * HIDF + f4;
    atomicAdd(o + 0, hv.x * norm);
    atomicAdd(o + 1, hv.y * norm);
    atomicAdd(o + 2, hv.z * norm);
    atomicAdd(o + 3, hv.w * norm);
}

// ---------------- self-loop + bias + leaky ----------------
// out = leaky(agg + h * dis^2 + bias)   (elementwise; out may alias agg)
__global__ void k_post(const float* __restrict__ agg, const float* __restrict__ h,
                       const float* __restrict__ dis, const float* __restrict__ bias,
                       float* __restrict__ out, int n_elems) {
    int t = blockIdx.x * blockDim.x + threadIdx.x;
    if (t >= n_elems) return;
    int i = t >> 6;            // node (HID=64)
    int f = t & (HIDF - 1);    // feature
    float s = dis[i];
    out[t] = leaky(agg[t] + h[t] * s * s + bias[f]);
}

extern "C" void kernel_launch(void* const* d_in, const int* in_sizes, int n_in,
                              void* d_out, int out_size, void* d_ws, size_t ws_size,
                              hipStream_t stream) {
    const float* x   = (const float*)d_in[0];
    const int*   ei  = (const int*)d_in[1];   // [2, E] int32 (JAX x64 disabled)
    const float* W1  = (const float*)d_in[2];
    const float* b1  = (const float*)d_in[3];
    const float* W2  = (const float*)d_in[4];
    const float* b2  = (const float*)d_in[5];
    const float* W3  = (const float*)d_in[6];
    const float* b3  = (const float*)d_in[7];
    const float* Wfc = (const float*)d_in[8];
    const float* bfc = (const float*)d_in[9];

    const int F_IN = 10;
    const int N = in_sizes[0] / F_IN;
    const int E = in_sizes[1] / 2;
    const int* src = ei;
    const int* dst = ei + E;

    // workspace: dis [N] | bufA [N*64] | bufB [N*64]
    float* dis  = (float*)d_ws;
    float* bufA = dis + (((size_t)N + 63) & ~(size_t)63);
    float* bufB = bufA + (size_t)N * HIDF;

    const int TB = 256;
    const int nh = N * HIDF;
    const int gemm_blocks = 256;              // 2048 waves grid-striding over 6250 tiles
    const int agg_blocks  = (int)(((long long)E * 16 + TB - 1) / TB);
    const int nh_blocks   = (nh + TB - 1) / TB;
    const int n_blocks    = (N + TB - 1) / TB;
    const int e_blocks    = (E + TB - 1) / TB;

    // normalization: deg -> deg^{-1/2}
    k_init_deg<<<n_blocks, TB, 0, stream>>>(dis, N);
    k_count_deg<<<e_blocks, TB, 0, stream>>>(dst, dis, E);
    k_rsqrt_inplace<<<n_blocks, TB, 0, stream>>>(dis, N);

    // ---- layer 1: h = x @ W1 (K=10, branchless tail) ----
    k_gemm_wmma<10, 64><<<gemm_blocks, TB, 0, stream>>>(x, W1, bfc, 0, bufB, N);
    k_zero<<<nh_blocks, TB, 0, stream>>>(bufA, nh);
    k_aggregate<<<agg_blocks, TB, 0, stream>>>(src, dst, dis, bufB, bufA, E);
    k_post<<<nh_blocks, TB, 0, stream>>>(bufA, bufB, dis, b1, bufA, nh);

    // ---- layer 2 ----
    k_gemm_wmma<64, 64><<<gemm_blocks, TB, 0, stream>>>(bufA, W2, bfc, 0, bufB, N);
    k_zero<<<nh_blocks, TB, 0, stream>>>(bufA, nh);
    k_aggregate<<<agg_blocks, TB, 0, stream>>>(src, dst, dis, bufB, bufA, E);
    k_post<<<nh_blocks, TB, 0, stream>>>(bufA, bufB, dis, b2, bufA, nh);

    // ---- layer 3 ----
    k_gemm_wmma<64, 64><<<gemm_blocks, TB, 0, stream>>>(bufA, W3, bfc, 0, bufB, N);
    k_zero<<<nh_blocks, TB, 0, stream>>>(bufA, nh);
    k_aggregate<<<agg_blocks, TB, 0, stream>>>(src, dst, dis, bufB, bufA, E);
    k_post<<<nh_blocks, TB, 0, stream>>>(bufA, bufB, dis, b3, bufA, nh);

    // ---- FC head: out = act3 @ Wfc + bfc  (NCOLS=4, masked stores only) ----
    k_gemm_wmma<64, 4><<<gemm_blocks, TB, 0, stream>>>(bufA, Wfc, bfc, 1,
                                                       (float*)d_out, N);
}